// GaussianEdgeGuideV2_61220463837596
// MI455X (gfx1250) — compile-verified
//
#include <hip/hip_runtime.h>

// GaussianEdgeGuideV2: edge-guided 3x3 softmax filtering.
// mask: (8,64,128,128) f32, edge: (8,1,64,64) f32 -> out (8,64,128,128) f32.
// Memory-bound: ~64MB HBM traffic -> ~2.8us at 23.3 TB/s. Weights depend only
// on (n,y,x) and are shared by the 64 channels; mask is streamed through LDS
// with double-buffered gfx1250 async global->LDS b128 copies overlapped with
// compute via s_wait_asynccnt thresholds.

#define GEG_N  8
#define GEG_C  64
#define GEG_H  128
#define GEG_W  128
#define GEG_EH 64
#define GEG_EW 64
#define GEG_THETA 10.0f

// LDS mask row: [0..3] left pad (halo at [3]), data [4..131], halo at [132].
// Data starts at float index 4 => every b128 deposit is 16-byte aligned.
#define GEG_ROW 136
#define GEG_CPS 8    // channels per stage
#define GEG_STAGES (GEG_C / GEG_CPS)

#if __has_builtin(__builtin_amdgcn_global_load_async_to_lds_b128)
  #define GEG_ASYNC 1
#else
  #define GEG_ASYNC 0
#endif

typedef int geg_v4i __attribute__((ext_vector_type(4)));
typedef __attribute__((address_space(3))) geg_v4i* geg_lds_v4i;

__device__ __forceinline__ void geg_copy16(const float* src, float* dst) {
#if GEG_ASYNC
  __builtin_amdgcn_global_load_async_to_lds_b128(
      (geg_v4i*)const_cast<float*>(src),   // global source (generic ptr)
      (geg_lds_v4i)dst,                    // LDS destination (addrspace(3))
      /*offset=*/0, /*cpol=*/0);
#else
  *(float4*)dst = *(const float4*)src;
#endif
}

__device__ __forceinline__ void geg_wait_async_3() {
#if GEG_ASYNC
#if __has_builtin(__builtin_amdgcn_s_wait_asynccnt)
  __builtin_amdgcn_s_wait_asynccnt(3);
#else
  asm volatile("s_wait_asynccnt 3" ::: "memory");
#endif
#endif
}
__device__ __forceinline__ void geg_wait_async_0() {
#if GEG_ASYNC
#if __has_builtin(__builtin_amdgcn_s_wait_asynccnt)
  __builtin_amdgcn_s_wait_asynccnt(0);
#else
  asm volatile("s_wait_asynccnt 0" ::: "memory");
#endif
#endif
}

// Zero page used as the source for out-of-range halo rows so that every wave
// issues the exact same number of async-load instructions each stage
// (EXEC-predicated async loads make ASYNCcnt thresholds unreliable).
__global__ void geg_zero_ws(float* z) { z[threadIdx.x] = 0.0f; }

__global__ __launch_bounds__(256)
void geg_weighted3x3(const float* __restrict__ mask,
                     const float* __restrict__ edge,
                     float* __restrict__ out,
                     const float* __restrict__ zrow) {
  const int blk = blockIdx.x;       // n*128 + y
  const int n   = blk >> 7;
  const int y   = blk & 127;
  const int tid = threadIdx.x;
  const int nb  = n * (GEG_C * GEG_H * GEG_W);

  __shared__ float e_lds[3][130];                     // upsampled edge rows
  __shared__ float wgt_lds[9][128];                   // per-pixel weights
  __shared__ float m_lds[2][GEG_CPS][3][GEG_ROW];     // double-buffered tiles

  // Issue one stage of async loads: 8 channels x 3 rows x 32 float4s = 768
  // b128 transfers, exactly 3 per thread, EXEC always all-ones.
  auto issue_stage = [&](int c0, int buf) {
#pragma unroll
    for (int j = 0; j < 3; ++j) {
      const int g  = j * 8 + (tid >> 5);   // (idx/32) in [0,24)
      const int cc = g / 3;
      const int r  = g % 3;
      const int xq = tid & 31;
      const int yy = y + r - 1;
      const bool valid = (yy >= 0) && (yy < GEG_H);
      const float* src = valid
          ? (mask + nb + ((c0 + cc) * GEG_H + yy) * GEG_W + xq * 4)
          : (zrow + xq * 4);
      geg_copy16(src, &m_lds[buf][cc][r][4 + xq * 4]);
    }
  };

  // Halo columns are never written by the loads; zero them once.
  for (int t = tid; t < 2 * GEG_CPS * 3; t += 256) {
    const int buf = t / (GEG_CPS * 3);
    const int rem = t % (GEG_CPS * 3);
    m_lds[buf][rem / 3][rem % 3][3]   = 0.0f;
    m_lds[buf][rem / 3][rem % 3][132] = 0.0f;
  }
  if (tid < 3) { e_lds[tid][0] = 0.0f; e_lds[tid][129] = 0.0f; }

  // Kick off stage 0 immediately; it overlaps the edge upsample + softmax.
  issue_stage(0, 0);

  // ---- Phase A: bilinear (align_corners) upsample 3 edge rows into LDS ----
  {
    const float s  = 63.0f / 127.0f;
    const float* eb = edge + n * (GEG_EH * GEG_EW);
    for (int t = tid; t < 3 * 128; t += 256) {
      const int r  = t >> 7;
      const int xx = t & 127;
      const int yy = y + r - 1;
      float v = 0.0f;
      if (yy >= 0 && yy < GEG_H) {
        const float ys = yy * s;
        const int   y0 = (int)ys;
        const float wy = ys - (float)y0;
        const int   y1 = min(y0 + 1, GEG_EH - 1);
        const float xs = xx * s;
        const int   x0 = (int)xs;
        const float wx = xs - (float)x0;
        const int   x1 = min(x0 + 1, GEG_EW - 1);
        const float tl = eb[y0 * GEG_EW + x0];
        const float tr = eb[y0 * GEG_EW + x1];
        const float bl = eb[y1 * GEG_EW + x0];
        const float br = eb[y1 * GEG_EW + x1];
        const float top = tl + (tr - tl) * wx;
        const float bot = bl + (br - bl) * wx;
        v = top + (bot - top) * wy;
      }
      e_lds[r][xx + 1] = v;
    }
  }
  __syncthreads();

  // ---- Phase B: softmax over 9 neighbor logits, shared by all channels ----
  if (tid < 128) {
    const int x = tid;
    const float ec = e_lds[1][x + 1];
    float sv[9];
    float sum = 0.0f;
#pragma unroll
    for (int k = 0; k < 9; ++k) {
      const int di = k / 3, dj = k % 3;
      const float le = ec - e_lds[di][x + dj];        // zero halo == padding
      const float v  = __expf(-GEG_THETA * le * le);  // logits <= 0, max == 0
      sv[k] = v;
      sum += v;
    }
    const float inv = 1.0f / sum;
#pragma unroll
    for (int k = 0; k < 9; ++k) wgt_lds[k][x] = sv[k] * inv;
  }
  __syncthreads();

  // Weights are loop-invariant: hoist the 9 values into registers.
  const int xi = tid & 127;
  const int cm = tid >> 7;   // 0 or 1: this thread covers channels cm+2q
  float w[9];
#pragma unroll
  for (int k = 0; k < 9; ++k) w[k] = wgt_lds[k][xi];

  // ---- Phase C: 8 stages x 8 channels, double-buffered async pipeline ----
  for (int it = 0; it < GEG_STAGES; ++it) {
    const int c0  = it * GEG_CPS;
    const int buf = it & 1;
    if (it + 1 < GEG_STAGES) {
      issue_stage(c0 + GEG_CPS, buf ^ 1);  // prefetch next stage
      geg_wait_async_3();                  // 3 newest (prefetch) may remain
    } else {
      geg_wait_async_0();
    }
    __syncthreads();

    float acc[4] = {0.0f, 0.0f, 0.0f, 0.0f};
#pragma unroll
    for (int di = 0; di < 3; ++di)
#pragma unroll
      for (int dj = 0; dj < 3; ++dj) {
        const float wk = w[di * 3 + dj];
#pragma unroll
        for (int q = 0; q < 4; ++q)
          acc[q] += m_lds[buf][cm + 2 * q][di][xi + dj + 3] * wk;
      }
#pragma unroll
    for (int q = 0; q < 4; ++q)
      out[nb + ((c0 + cm + 2 * q) * GEG_H + y) * GEG_W + xi] = acc[q];

    __syncthreads();  // all reads of this buffer done before it is refilled
  }
}

extern "C" void kernel_launch(void* const* d_in, const int* in_sizes, int n_in,
                              void* d_out, int out_size, void* d_ws, size_t ws_size,
                              hipStream_t stream) {
  (void)in_sizes; (void)n_in; (void)ws_size; (void)out_size;
  const float* mask = (const float*)d_in[0];
  const float* edge = (const float*)d_in[1];
  float* outp = (float*)d_out;
  float* zrow = (float*)d_ws;

  // Zero page for halo-row sources (ws is poisoned by the harness; zero it
  // every call -- deterministic and graph-capture safe).
  hipLaunchKernelGGL(geg_zero_ws, dim3(1), dim3(128), 0, stream, zrow);

  dim3 grid(GEG_N * GEG_H);   // one block per (n, y) row
  dim3 block(256);
  hipLaunchKernelGGL(geg_weighted3x3, grid, block, 0, stream,
                     mask, edge, outp, (const float*)zrow);
}